// VolumetricSampler_67551245631715
// MI455X (gfx1250) — compile-verified
//
#include <hip/hip_runtime.h>

// ---- constants mirroring the reference ----
#define N_RAYS     32768
#define MAX_S      256
#define GRID_RES   128
#define STEP       0.01f
#define NEAR_PLANE 0.05f
#define FAR_PLANE  1e10f

// alpha > 0.01  <=>  1-exp(-sigma*STEP) > 0.01  <=>  sigma > -ln(0.99)/STEP
#define SIGMA_THRE 1.0050335853f

typedef float v4f __attribute__((ext_vector_type(4)));

// 4 samples per thread -> 64 threads per ray. A wave32 is ray-uniform and
// covers 128 consecutive samples (high gather locality in L0/L2).
// Outputs: 5 planar arrays of N_RAYS*MAX_S floats, written as b128 NT stores.
__global__ __launch_bounds__(256) void vol_sampler_kernel(
    const float* __restrict__ rays_o,
    const float* __restrict__ rays_d,
    const int*   __restrict__ occ_grid,
    const float* __restrict__ density_grid,
    float*       __restrict__ out)
{
    const long long NS = (long long)N_RAYS * MAX_S;

    const int gid = blockIdx.x * blockDim.x + threadIdx.x;
    const int ray = gid >> 6;          // 64 threads per ray
    const int s0  = (gid & 63) << 2;   // first of 4 samples
    if (ray >= N_RAYS) return;

    // ray index is uniform across each wave32: pin it to an SGPR so the ray
    // loads take the scalar-base (saddr) path instead of per-lane 64b address
    // arithmetic + broadcast VMEM.
    const int ray_u = __builtin_amdgcn_readfirstlane(ray);

    const float ox = rays_o[ray_u * 3 + 0];
    const float oy = rays_o[ray_u * 3 + 1];
    const float oz = rays_o[ray_u * 3 + 2];
    const float dx = rays_d[ray_u * 3 + 0];
    const float dy = rays_d[ray_u * 3 + 1];
    const float dz = rays_d[ray_u * 3 + 2];

    // ---- slab test (d_safe = where(|d|<1e-8, 1e-8, d)) ----
    // v_rcp_f32 (1 ulp) instead of the IEEE div_scale/div_fixup chain: the
    // result only feeds threshold comparisons.
    const float sx = (fabsf(dx) < 1e-8f) ? 1e-8f : dx;
    const float sy = (fabsf(dy) < 1e-8f) ? 1e-8f : dy;
    const float sz = (fabsf(dz) < 1e-8f) ? 1e-8f : dz;
    const float ivx = __builtin_amdgcn_rcpf(sx);
    const float ivy = __builtin_amdgcn_rcpf(sy);
    const float ivz = __builtin_amdgcn_rcpf(sz);

    const float t1x = (-1.0f - ox) * ivx, t2x = (1.0f - ox) * ivx;
    const float t1y = (-1.0f - oy) * ivy, t2y = (1.0f - oy) * ivy;
    const float t1z = (-1.0f - oz) * ivz, t2z = (1.0f - oz) * ivz;

    float t_near = fmaxf(fmaxf(fminf(t1x, t2x), fminf(t1y, t2y)), fminf(t1z, t2z));
    float t_far  = fminf(fminf(fmaxf(t1x, t2x), fmaxf(t1y, t2y)), fmaxf(t1z, t2z));
    t_near = fmaxf(t_near, NEAR_PLANE);
    t_far  = fminf(t_far,  FAR_PLANE);
    const bool ray_valid = (t_near < t_far);

    // gfx1250 prefetch of the grid lines the first sample will gather from
    {
        const float tm0 = t_near + ((float)s0 + 0.5f) * STEP;
        const float fx = fminf(fmaxf((ox + dx * tm0 + 1.0f) * 64.0f, 0.0f), 127.0f);
        const float fy = fminf(fmaxf((oy + dy * tm0 + 1.0f) * 64.0f, 0.0f), 127.0f);
        const float fz = fminf(fmaxf((oz + dz * tm0 + 1.0f) * 64.0f, 0.0f), 127.0f);
        const int lin0 = ((int)fx * GRID_RES + (int)fy) * GRID_RES + (int)fz;
        __builtin_prefetch(&density_grid[lin0], 0, 1);   // -> global_prefetch_b8
        __builtin_prefetch(&occ_grid[lin0],     0, 1);
    }

    float st[4], en[4], sg[4], mk[4];
#pragma unroll
    for (int j = 0; j < 4; ++j) {
        const float fi      = (float)(s0 + j);
        const float t_start = t_near + fi * STEP;
        const float t_end   = t_start + STEP;
        const bool  in_rng  = (t_end <= t_far);
        const float t_mid   = 0.5f * (t_start + t_end);

        const float px = ox + dx * t_mid;
        const float py = oy + dy * t_mid;
        const float pz = oz + dz * t_mid;

        // idx = clip(floor((p+1)*res/2), 0, res-1). After the clamp the value
        // is in [0,127], so truncating v_cvt_i32_f32 == floor (no v_floor).
        const float fx = fminf(fmaxf((px + 1.0f) * 64.0f, 0.0f), 127.0f);
        const float fy = fminf(fmaxf((py + 1.0f) * 64.0f, 0.0f), 127.0f);
        const float fz = fminf(fmaxf((pz + 1.0f) * 64.0f, 0.0f), 127.0f);
        const int ix = (int)fx;
        const int iy = (int)fy;
        const int iz = (int)fz;
        const int lin = (ix * GRID_RES + iy) * GRID_RES + iz;

        const bool  occ   = occ_grid[lin] > 0;
        const float sigma = density_grid[lin];

        // alpha > ALPHA_THRE strength-reduced to sigma > -ln(0.99)/STEP
        const bool m = ray_valid && in_rng && occ && (sigma > SIGMA_THRE);
        st[j] = m ? t_start : 0.0f;
        en[j] = m ? t_end   : 0.0f;
        sg[j] = m ? sigma   : 0.0f;
        mk[j] = m ? 1.0f    : 0.0f;
    }

    const long long base = (long long)ray_u * MAX_S + s0;  // 16B aligned
    const float rid = (float)ray_u;                        // ray_indices, unmasked

    v4f vst = { st[0], st[1], st[2], st[3] };
    v4f ven = { en[0], en[1], en[2], en[3] };
    v4f vsg = { sg[0], sg[1], sg[2], sg[3] };
    v4f vid = { rid,   rid,   rid,   rid   };
    v4f vmk = { mk[0], mk[1], mk[2], mk[3] };

    // write-once 168MB output stream: b128 stores with NT temporal hint so the
    // stream does not evict the 16MB of grid data resident in L2.
    __builtin_nontemporal_store(vst, (v4f*)(out + 0 * NS + base));
    __builtin_nontemporal_store(ven, (v4f*)(out + 1 * NS + base));
    __builtin_nontemporal_store(vsg, (v4f*)(out + 2 * NS + base));
    __builtin_nontemporal_store(vid, (v4f*)(out + 3 * NS + base));
    __builtin_nontemporal_store(vmk, (v4f*)(out + 4 * NS + base));
}

extern "C" void kernel_launch(void* const* d_in, const int* in_sizes, int n_in,
                              void* d_out, int out_size, void* d_ws, size_t ws_size,
                              hipStream_t stream) {
    const float* rays_o       = (const float*)d_in[0];
    const float* rays_d       = (const float*)d_in[1];
    const int*   occ_grid     = (const int*)d_in[2];
    const float* density_grid = (const float*)d_in[3];
    float*       out          = (float*)d_out;

    const int total_threads = N_RAYS * (MAX_S / 4);   // 2,097,152
    const int block = 256;                            // 8 wave32s
    const int grid  = (total_threads + block - 1) / block;

    vol_sampler_kernel<<<grid, block, 0, stream>>>(rays_o, rays_d, occ_grid,
                                                   density_grid, out);
}